// LightGCN_37022618091924
// MI455X (gfx1250) — compile-verified
//
#include <hip/hip_runtime.h>

// ---- problem constants (from reference) ----
#define N_USERS_C 200000
#define N_ITEMS_C 50000
#define N_TOTAL_C 250000
#define EMB_C     64
#define N_LAYERS_C 3

typedef __attribute__((ext_vector_type(2))) float v2f;
typedef __attribute__((ext_vector_type(8))) float v8f;

// ---------------------------------------------------------------------------
// Init: cur = acc = concat(user_emb, item_emb); next = 0.   float4 vectorized.
// Boundary (200000*64 = 12.8M floats) is float4-aligned, no straddle.
// ---------------------------------------------------------------------------
__global__ void lgcn_init(const float* __restrict__ ue, const float* __restrict__ ie,
                          float* __restrict__ cur, float* __restrict__ acc,
                          float* __restrict__ nxt, int nVec4, int userVec4) {
    int i = blockIdx.x * blockDim.x + threadIdx.x;
    if (i >= nVec4) return;
    float4 v = (i < userVec4) ? ((const float4*)ue)[i]
                              : ((const float4*)ie)[i - userVec4];
    ((float4*)cur)[i] = v;
    ((float4*)acc)[i] = v;
    ((float4*)nxt)[i] = make_float4(0.f, 0.f, 0.f, 0.f);
}

// ---------------------------------------------------------------------------
// SpMM: one wave32 per edge. Lane l handles the float2 at column 2l of the
// 64-wide row: a single coalesced 256B gather from cur[col] (L2-resident),
// then 2 hardware f32 atomic adds into next[row] (also L2-resident).
// Edge list is streamed from HBM; prefetch a few iterations ahead.
// ---------------------------------------------------------------------------
__global__ void lgcn_spmm(const float* __restrict__ cur, float* __restrict__ nxt,
                          const int* __restrict__ rows, const int* __restrict__ cols,
                          const float* __restrict__ vals, int nnz) {
    const int lane   = threadIdx.x & 31;
    const int wave   = blockIdx.x * (blockDim.x >> 5) + (threadIdx.x >> 5);
    const int stride = gridDim.x * (blockDim.x >> 5);
    for (int e = wave; e < nnz; e += stride) {
        if (lane == 0 && (e + 4 * stride) < nnz) {
            __builtin_prefetch(rows + e + 4 * stride, 0, 0);   // global_prefetch_b8
            __builtin_prefetch(cols + e + 4 * stride, 0, 0);
            __builtin_prefetch(vals + e + 4 * stride, 0, 0);
        }
        const int   r = rows[e];
        const int   c = cols[e];
        const float v = vals[e];
        const float2 x = ((const float2*)(cur + (size_t)c * EMB_C))[lane];
        float* dst = nxt + (size_t)r * EMB_C + 2 * lane;
        unsafeAtomicAdd(dst,     v * x.x);   // global_atomic_add_f32 (no CAS loop)
        unsafeAtomicAdd(dst + 1, v * x.y);
    }
}

// ---------------------------------------------------------------------------
// acc += next, and zero the buffer that becomes next layer's target.
// ---------------------------------------------------------------------------
__global__ void lgcn_accum(float* __restrict__ acc, const float* __restrict__ nxt,
                           float* __restrict__ zero_me, int nVec4) {
    int i = blockIdx.x * blockDim.x + threadIdx.x;
    if (i >= nVec4) return;
    float4 a = ((float4*)acc)[i];
    float4 n = ((const float4*)nxt)[i];
    a.x += n.x; a.y += n.y; a.z += n.z; a.w += n.w;
    ((float4*)acc)[i]     = a;
    ((float4*)zero_me)[i] = make_float4(0.f, 0.f, 0.f, 0.f);
}

// ---------------------------------------------------------------------------
// Final batched dot products via V_WMMA_F32_16X16X4_F32 (exact fp32 math).
// One wave handles 16 pairs: D = U(16x64) x IT^T(64x16) accumulated over 16
// K=4 chunks; we keep the diagonal and scale by 1/16 (= (1/4)^2 layer avg).
//
// Layouts (ISA 7.12.2, 32-bit):
//   A 16x4 : lane -> M = lane%16, K = 2*(lane/16)+vgpr
//   B 4x16 : lane -> N = lane%16, K = 2*(lane/16)+vgpr
//   C/D    : lane -> N = lane%16, vgpr v -> M = v + 8*(lane/16)
//   diag p<8  -> lane p,     c[p]
//   diag p>=8 -> lane p+16,  c[p-8]
// ---------------------------------------------------------------------------
__global__ void lgcn_dot_wmma(const float* __restrict__ acc,
                              const int* __restrict__ batch,
                              float* __restrict__ out, int nPairs) {
    const int lane = threadIdx.x & 31;
    const int wave = blockIdx.x * (blockDim.x >> 5) + (threadIdx.x >> 5);
    const int g0   = wave * 16;
    if (g0 >= nPairs) return;

    const int m  = lane & 15;        // A-row index == B-col index for this lane
    const int kh = lane >> 4;        // K half-select

    const int p  = g0 + m;
    const long long urow = (long long)batch[2 * p + 0];
    const long long irow = (long long)N_USERS_C + (long long)batch[2 * p + 1];
    const float* uPtr = acc + urow * EMB_C + 2 * kh;
    const float* iPtr = acc + irow * EMB_C + 2 * kh;

    v8f c = {};
#pragma unroll
    for (int kc = 0; kc < 16; ++kc) {
        v2f a = *(const v2f*)(uPtr + 4 * kc);
        v2f b = *(const v2f*)(iPtr + 4 * kc);
        // 8 args: (neg_a, A, neg_b, B, c_mod, C, reuse_a, reuse_b)
        c = __builtin_amdgcn_wmma_f32_16x16x4_f32(false, a, false, b,
                                                  (short)0, c, false, false);
    }

    // Diagonal extraction + 1/16 scale (acc/4 on both sides of the dot).
    if (lane < 8) {
        int q = g0 + lane;
        if (q < nPairs) out[q] = c[lane] * 0.0625f;
    } else if (lane >= 24) {
        int q = g0 + (lane - 16);
        if (q < nPairs) out[q] = c[lane - 24] * 0.0625f;
    }
}

// ---------------------------------------------------------------------------
// Host launcher. d_in: user_emb, item_emb, a_rows, a_cols, a_vals, batch.
// d_ws: 3 x (250000*64) floats = 192 MB  (cur | next | acc).
// ---------------------------------------------------------------------------
extern "C" void kernel_launch(void* const* d_in, const int* in_sizes, int n_in,
                              void* d_out, int out_size, void* d_ws, size_t ws_size,
                              hipStream_t stream) {
    const float* user_emb = (const float*)d_in[0];
    const float* item_emb = (const float*)d_in[1];
    const int*   a_rows   = (const int*)d_in[2];
    const int*   a_cols   = (const int*)d_in[3];
    const float* a_vals   = (const float*)d_in[4];
    const int*   batch    = (const int*)d_in[5];
    float*       out      = (float*)d_out;

    const int nnz    = in_sizes[2];
    const int nPairs = in_sizes[5] / 2;

    const int nElems   = N_TOTAL_C * EMB_C;        // 16,000,000 floats (64 MB)
    const int nVec4    = nElems / 4;               // 4,000,000
    const int userVec4 = (N_USERS_C * EMB_C) / 4;  // 3,200,000

    float* w0  = (float*)d_ws;        // cur
    float* w1  = w0 + nElems;         // next
    float* acc = w1 + nElems;         // accumulator

    // init: cur = acc = emb, next = 0
    {
        int threads = 256;
        int blocks  = (nVec4 + threads - 1) / threads;
        lgcn_init<<<blocks, threads, 0, stream>>>(user_emb, item_emb,
                                                  w0, acc, w1, nVec4, userVec4);
    }

    float* cur = w0;
    float* nxt = w1;
    for (int layer = 0; layer < N_LAYERS_C; ++layer) {
        {   // SpMM: cur -> nxt
            int threads = 256;                     // 8 waves/block
            int blocks  = 4096;                    // 32768 waves, grid-stride
            lgcn_spmm<<<blocks, threads, 0, stream>>>(cur, nxt, a_rows, a_cols,
                                                      a_vals, nnz);
        }
        {   // acc += nxt; zero old cur (it becomes next layer's target)
            int threads = 256;
            int blocks  = (nVec4 + threads - 1) / threads;
            lgcn_accum<<<blocks, threads, 0, stream>>>(acc, nxt, cur, nVec4);
        }
        float* t = cur; cur = nxt; nxt = t;        // swap
    }

    // final batched dot products (WMMA f32)
    {
        int threads = 256;                                   // 8 waves/block
        int waves   = (nPairs + 15) / 16;
        int blocks  = (waves + 7) / 8;
        lgcn_dot_wmma<<<blocks, threads, 0, stream>>>(acc, batch, out, nPairs);
    }
}